// GHMRLoss_55654186222056
// MI455X (gfx1250) — compile-verified
//
#include <hip/hip_runtime.h>

// GHM-R loss for MI455X (gfx1250, wave32).
// Memory-bound streaming reduction: 201 MB in -> ~8.6 us floor @ 23.3 TB/s.
// Per-thread privatized LDS histogram (bank-conflict-free b64 RMW, stride 22),
// block-level reduction done with V_WMMA_F32_16X16X4_F32 (A=ones), then a
// deterministic two-stage inter-block combine through d_ws.

typedef float v2f __attribute__((ext_vector_type(2)));
typedef float v8f __attribute__((ext_vector_type(8)));

#define TPB   256
#define ROWF  22   // floats per thread-row in LDS: 10*(loss,count) + tot + zero pad
                   // stride 22 -> t*22 mod 64 gives 32 distinct even residues
                   // => conflict-free ds_load_b64 / ds_store_b64 per wave.

__device__ __forceinline__ void ghmr_acc(float x, float y, int m,
                                         float2* rowp, float& tot) {
  const float mu  = 0.02f;
  const float mu2 = mu * mu;
  float d    = x - y;
  float s    = sqrtf(fmaf(d, d, mu2));
  float loss = s - mu;
  float g    = fabsf(d) / s;            // in [0,1)
  int idx = (int)(g * 10.0f);
  idx = idx < 0 ? 0 : (idx > 9 ? 9 : idx);
  float vf = (m > 0) ? 1.0f : 0.0f;
  float2 cur = rowp[idx];               // ds_load_b64 (private row, no races)
  cur.x = fmaf(loss, vf, cur.x);        // per-bin loss sum (valid only)
  cur.y += vf;                          // per-bin valid count
  rowp[idx] = cur;                      // ds_store_b64
  tot += (float)m;                      // sum(mask) in register
}

__global__ __launch_bounds__(TPB) void ghmr_main(
    const float* __restrict__ inp, const float* __restrict__ tgt,
    const int* __restrict__ msk, float* __restrict__ ws,
    int n4, int nblk) {
  __shared__ float smem[TPB * ROWF];
  const int tid = threadIdx.x;
  float* row = smem + tid * ROWF;
#pragma unroll
  for (int i = 0; i < ROWF; ++i) row[i] = 0.0f;
  float2* rowp = (float2*)row;          // 8B aligned: ROWF even
  float tot = 0.0f;

  const float4* in4 = (const float4*)inp;   // global_load_b128
  const float4* tg4 = (const float4*)tgt;
  const int4*   mk4 = (const int4*)msk;
  const int stride = gridDim.x * TPB;
  for (int i = blockIdx.x * TPB + tid; i < n4; i += stride) {
    float4 a = in4[i];
    float4 b = tg4[i];
    int4   m = mk4[i];
    ghmr_acc(a.x, b.x, m.x, rowp, tot);
    ghmr_acc(a.y, b.y, m.y, rowp, tot);
    ghmr_acc(a.z, b.z, m.z, rowp, tot);
    ghmr_acc(a.w, b.w, m.w, rowp, tot);
  }
  row[20] = tot;                        // column 20 = sum(mask); column 21 stays 0
  __syncthreads();

  // ---- Block reduction of the 256x21 partial matrix with WMMA ----
  // D = ones(16x4) x B(4x16) + C  sums 4 thread-rows per issue; 64 chunks
  // cover all 256 rows; every row of D holds the column sums.
  // Wave 0 only: threads 0..31 == one full wave32 -> EXEC all ones at WMMA.
  if (tid < 32) {
    const int lane = tid;
    const int n    = lane & 15;           // B column this lane holds
    const int koff = (lane >> 4) << 1;    // VGPR0 holds K=0 (lo lanes) / K=2 (hi)
    v2f ones; ones[0] = 1.0f; ones[1] = 1.0f;
#pragma unroll
    for (int G = 0; G < 2; ++G) {         // two 16-column groups cover 21 cols
      const int col = (G << 4) + n;
      const int cc  = col < 21 ? col : 21;  // pad column 21 is zero
      v8f c = {};
      for (int j = 0; j < 64; ++j) {
        const int r = (j << 2) + koff;    // rows 4j+{0,1} (lo) / 4j+{2,3} (hi)
        v2f bb;
        bb[0] = smem[r * ROWF + cc];
        bb[1] = smem[(r + 1) * ROWF + cc];
        c = __builtin_amdgcn_wmma_f32_16x16x4_f32(
            /*neg_a=*/false, ones, /*neg_b=*/false, bb,
            /*c_mod=*/(short)0, c, /*reuse_a=*/false, /*reuse_b=*/false);
      }
      if (lane < 16 && col < 21)          // D row M=0 lives in c[0], lanes 0..15
        ws[col * nblk + blockIdx.x] = c[0];
    }
  }
}

// Deterministic stage-2: reduce 21 columns over nblk blocks, apply GHM weights.
__global__ __launch_bounds__(TPB) void ghmr_finalize(
    const float* __restrict__ ws, int nblk, float* __restrict__ out) {
  __shared__ float red[TPB];
  __shared__ float col[21];
  const int tid = threadIdx.x;
  for (int c = 0; c < 21; ++c) {
    float p = 0.0f;
    for (int i = tid; i < nblk; i += TPB) p += ws[c * nblk + i];
    red[tid] = p;
    __syncthreads();
    for (int s = TPB >> 1; s > 0; s >>= 1) {
      if (tid < s) red[tid] += red[tid + s];
      __syncthreads();
    }
    if (tid == 0) col[c] = red[0];
    __syncthreads();
  }
  if (tid == 0) {
    float tot = fmaxf(col[20], 1.0f);                 // max(sum(mask),1)
    float nb = 0.0f;                                  // # non-empty bins
    for (int b = 0; b < 10; ++b) nb += (col[2 * b + 1] > 0.0f) ? 1.0f : 0.0f;
    float nm = fmaxf(nb, 1.0f);
    float s = 0.0f;
    for (int b = 0; b < 10; ++b) {
      float cnt = col[2 * b + 1];
      float w = (cnt > 0.0f) ? (tot / fmaxf(cnt, 1.0f)) : 0.0f;
      s += col[2 * b] * (w / nm);                     // sum(loss * per_bin_w)
    }
    out[0] = (s / tot) * 1.0f;                        // LOSS_WEIGHT = 1.0
  }
}

extern "C" void kernel_launch(void* const* d_in, const int* in_sizes, int n_in,
                              void* d_out, int out_size, void* d_ws, size_t ws_size,
                              hipStream_t stream) {
  const float* inp = (const float*)d_in[0];
  const float* tgt = (const float*)d_in[1];
  const int*   msk = (const int*)d_in[2];
  float* out = (float*)d_out;
  float* ws  = (float*)d_ws;

  const int n  = in_sizes[0];           // 16,777,216 (divisible by 4)
  const int n4 = n >> 2;

  int nblk = 1024;                                        // ~32 waves/WGP territory
  const int maxblk_n = (n4 + TPB - 1) / TPB;
  if (nblk > maxblk_n) nblk = maxblk_n;
  const long long maxblk_ws = (long long)(ws_size / (21 * sizeof(float)));
  if ((long long)nblk > maxblk_ws) nblk = (int)maxblk_ws;
  if (nblk < 1) nblk = 1;

  ghmr_main<<<nblk, TPB, 0, stream>>>(inp, tgt, msk, ws, n4, nblk);
  ghmr_finalize<<<1, TPB, 0, stream>>>(ws, nblk, out);
}